// Retrace_22325240004684
// MI455X (gfx1250) — compile-verified
//
#include <hip/hip_runtime.h>
#include <hip/hip_bf16.h>

// Retrace MSE, MI455X (gfx1250).
// Memory-bound (~100 MB @ 23.3 TB/s => ~4.3us floor). Wave-per-row, wave32
// affine suffix-scan for the backward recurrence, CDNA5 async global->LDS
// double buffering (global_load_async_to_lds_b32 / s_wait_asynccnt).

#define GAMMA 0.99f
#define CLIP_EPS 1e-10f

constexpr int TP1     = 1025;        // columns per row (T+1)
constexpr int TSTEPS  = 1024;        // recurrence steps t = 1..1024
constexpr int WAVES   = 8;           // waves per block
constexpr int BLOCK   = WAVES * 32;  // 256 threads
constexpr int NCHUNKS = TSTEPS / 32; // 32 chunks of 32 timesteps

// ---- CDNA5 async global->LDS helpers (inline asm; ASYNCcnt-tracked) --------

__device__ __forceinline__ unsigned lds_addr(const void* p) {
    // Generic pointer to LDS: low 32 bits are the LDS byte offset
    // (flat-address LDS aperture mapping, CDNA5 ISA 10.2).
    return (unsigned)(unsigned long long)p;
}

__device__ __forceinline__ void async_load_f32(unsigned lds_byte_addr,
                                               const float* gaddr) {
    // each lane copies 4B from its global address in vaddr to its LDS address in vdst.
    asm volatile("global_load_async_to_lds_b32 %0, %1, off"
                 :: "v"(lds_byte_addr), "v"(gaddr)
                 : "memory");
}

__device__ __forceinline__ void wait_async_le6() {
    asm volatile("s_wait_asynccnt 0x6" ::: "memory");
}
__device__ __forceinline__ void wait_async_0() {
    asm volatile("s_wait_asynccnt 0x0" ::: "memory");
}

// ---- main kernel: one wave32 per batch row ---------------------------------

__global__ __launch_bounds__(BLOCK)
void retrace_mse_kernel(const float* __restrict__ Q,
                        const float* __restrict__ eQ,
                        const float* __restrict__ tQ,
                        const float* __restrict__ RW,
                        const float* __restrict__ TPP,
                        const float* __restrict__ BPP,
                        float* __restrict__ partial)
{
    // per-wave double-buffered staging: [wave][buf][stream][lane]
    __shared__ float smem[WAVES][2][6][32];
    __shared__ float wsum[WAVES];

    const int lane = threadIdx.x & 31;
    const int w    = threadIdx.x >> 5;
    const int row  = blockIdx.x * WAVES + w;

    const float* Rq = Q   + (size_t)row * TP1;
    const float* Rr = RW  + (size_t)row * TP1;
    const float* Re = eQ  + (size_t)row * TP1;
    const float* Rt = tQ  + (size_t)row * TP1;
    const float* Rp = TPP + (size_t)row * TP1;
    const float* Rb = BPP + (size_t)row * TP1;

    const float q_last = Rq[TSTEPS];  // Q[row, 1024] (bootstrap value)

    auto issue = [&](int c, int bufi) {
        // chunk c covers t in [1+32c, 32+32c]; lane l <-> t = 1+32c+l
        const int t   = 1 + c * 32 + lane;
        const int tm1 = t - 1;
        int tp1c = t + 1; if (tp1c > TSTEPS) tp1c = TSTEPS;  // clamp (lane t==1024 value is overridden)
        float* buf = &smem[w][bufi][0][0];
        async_load_f32(lds_addr(&buf[0 * 32 + lane]), Rq + tm1);   // Q[t-1]
        async_load_f32(lds_addr(&buf[1 * 32 + lane]), Rr + t);     // r[t]
        async_load_f32(lds_addr(&buf[2 * 32 + lane]), Re + tp1c);  // eQ[t+1]
        async_load_f32(lds_addr(&buf[3 * 32 + lane]), Rt + tp1c);  // tQ[t+1]
        async_load_f32(lds_addr(&buf[4 * 32 + lane]), Rp + tp1c);  // pi[t+1]
        async_load_f32(lds_addr(&buf[5 * 32 + lane]), Rb + tp1c);  // mu[t+1]
    };

    // prologue: stage highest-t chunk
    issue(NCHUNKS - 1, (NCHUNKS - 1) & 1);

    float acc   = 0.0f;  // per-lane sum of squared errors
    float carry = 0.0f;  // z_{t0+32}; don't-care for first chunk (b_1024 == 0)

    for (int c = NCHUNKS - 1; c >= 0; --c) {
        if (c > 0) { issue(c - 1, (c - 1) & 1); wait_async_le6(); }
        else       { wait_async_0(); }

        const float* buf = &smem[w][c & 1][0][0];
        const float qm1  = buf[0 * 32 + lane];
        const float rt   = buf[1 * 32 + lane];
        const float e    = buf[2 * 32 + lane];
        const float tqv  = buf[3 * 32 + lane];
        const float tpv  = buf[4 * 32 + lane];
        const float bpv  = buf[5 * 32 + lane];

        // a_t = r_t + g*(eQ_{t+1} - c_{t+1}*tQ_{t+1}),  b_t = g*c_{t+1}
        float cc = fminf(fmaxf(tpv / bpv, CLIP_EPS), 1.0f);
        float a  = fmaf(GAMMA, fmaf(-cc, tqv, e), rt);
        float b  = GAMMA * cc;
        const int t = 1 + c * 32 + lane;
        if (t == TSTEPS) { a = q_last; b = 0.0f; }  // z_1024 = Q[:,1024]

        // wave32 suffix scan of affine maps: F_l = f_l o f_{l+1} o ... o f_31
        float A = a, Bm = b;
        #pragma unroll
        for (int d = 1; d < 32; d <<= 1) {
            float A2 = __shfl_down(A,  d, 32);
            float B2 = __shfl_down(Bm, d, 32);
            if (lane + d < 32) { A = fmaf(Bm, A2, A); Bm = Bm * B2; }
        }
        float z  = fmaf(Bm, carry, A);     // z_t
        float dz = qm1 - z;                // Q[:,t-1] - Q_ret
        acc = fmaf(dz, dz, acc);
        carry = __shfl(z, 0, 32);          // z_{t0} feeds next (lower-t) chunk
    }

    // wave reduce (deterministic fixed tree)
    #pragma unroll
    for (int d = 16; d >= 1; d >>= 1) acc += __shfl_down(acc, d, 32);
    if (lane == 0) wsum[w] = acc;
    __syncthreads();
    if (threadIdx.x == 0) {
        float s = 0.0f;
        #pragma unroll
        for (int i = 0; i < WAVES; ++i) s += wsum[i];
        partial[blockIdx.x] = s;
    }
}

// ---- final reduction: single block, fixed-order => deterministic -----------

__global__ __launch_bounds__(256)
void final_reduce_kernel(const float* __restrict__ partial,
                         float* __restrict__ out, int n, float scale)
{
    __shared__ float sm[256];
    float s = 0.0f;
    for (int i = threadIdx.x; i < n; i += 256) s += partial[i];
    sm[threadIdx.x] = s;
    __syncthreads();
    #pragma unroll
    for (int d = 128; d >= 1; d >>= 1) {
        if (threadIdx.x < d) sm[threadIdx.x] += sm[threadIdx.x + d];
        __syncthreads();
    }
    if (threadIdx.x == 0) out[0] = sm[0] * scale;
}

extern "C" void kernel_launch(void* const* d_in, const int* in_sizes, int n_in,
                              void* d_out, int out_size, void* d_ws, size_t ws_size,
                              hipStream_t stream) {
    // setup_inputs order: Q, expected_target_Q, target_Q, rewards,
    //                     target_policy_probs, behaviour_policy_probs
    const float* Q  = (const float*)d_in[0];
    const float* eQ = (const float*)d_in[1];
    const float* tQ = (const float*)d_in[2];
    const float* RW = (const float*)d_in[3];
    const float* TP = (const float*)d_in[4];
    const float* BP = (const float*)d_in[5];

    const int B       = in_sizes[0] / TP1;  // 4096
    const int nblocks = B / WAVES;          // 512

    float* partial = (float*)d_ws;          // 512 floats of scratch

    retrace_mse_kernel<<<nblocks, BLOCK, 0, stream>>>(Q, eQ, tQ, RW, TP, BP, partial);

    const float scale = 1.0f / ((float)B * (float)TSTEPS);
    final_reduce_kernel<<<1, 256, 0, stream>>>(partial, (float*)d_out, nblocks, scale);
}